// HighOrderFunction_18502719111479
// MI455X (gfx1250) — compile-verified
//
#include <hip/hip_runtime.h>

// Problem constants (fixed by the reference)
#define Bc 8
#define Cc 64
#define Hc 256
#define Wc 256
#define TH 8     // tile rows per block
#define TW 32    // tile cols per block
// weights: [25][B][1][H][W], x: [B][C][H][W], out: [B][C][H][W]

typedef int   v4i __attribute__((ext_vector_type(4)));
typedef float v4f __attribute__((ext_vector_type(4)));
typedef __attribute__((address_space(1))) v4i* gv4i_ptr;   // global
typedef __attribute__((address_space(3))) v4i* lv4i_ptr;   // LDS

// ---- CDNA5 async global->LDS 16-byte copy (ASYNCcnt path) ----
__device__ __forceinline__ void async_copy_b128(const float* g, float* l) {
#if __has_builtin(__builtin_amdgcn_global_load_async_to_lds_b128)
    __builtin_amdgcn_global_load_async_to_lds_b128((gv4i_ptr)g, (lv4i_ptr)l, 0, 0);
#else
    unsigned ldsoff = (unsigned)(unsigned long long)(lv4i_ptr)l;
    unsigned long long ga = (unsigned long long)g;
    asm volatile("global_load_async_to_lds_b128 %0, %1, off"
                 :: "v"(ldsoff), "v"(ga)
                 : "memory");
#endif
}

__device__ __forceinline__ void wait_async_zero() {
#if __has_builtin(__builtin_amdgcn_s_wait_asynccnt)
    __builtin_amdgcn_s_wait_asynccnt(0);
#else
    asm volatile("s_wait_asynccnt 0" ::: "memory");
#endif
}

__global__ __launch_bounds__(256)
void dynconv5x5_kernel(const float* __restrict__ x,
                       const float* __restrict__ wts,
                       float* __restrict__ out) {
    __shared__ float wlds[25 * TH * TW];   // 25.6 KB weight tile, reused by 64 channels

    const int b  = blockIdx.z;
    const int h0 = blockIdx.y * TH;
    const int w0 = blockIdx.x * TW;
    const int t  = threadIdx.x;
    const size_t plane = (size_t)Hc * Wc;

    // ---------- Phase 1: async-stage weight tile [25][TH][TW] into LDS ----------
    // 25*TH*TW*4 B = 25600 B = 1600 chunks of 16 B; 64 chunks per tap k.
    for (int c = t; c < 25 * TH * TW / 4; c += 256) {
        const int k   = c >> 6;          // tap index
        const int rem = c & 63;
        const int r   = rem >> 3;        // row in tile
        const int cw  = (rem & 7) << 2;  // col (multiple of 4)
        const float* g = wts + ((size_t)k * Bc + b) * plane
                             + (size_t)(h0 + r) * Wc + (w0 + cw);
        async_copy_b128(g, &wlds[(k * TH + r) * TW + cw]);
    }
    wait_async_zero();
    __syncthreads();

    // ---------- Phase 2: compute ----------
    const int c = t & 63;        // channel
    const int q = t >> 6;        // 0..3 -> rows 2q, 2q+1 of the tile
    const float* xc = x   + ((size_t)b * Cc + c) * plane;
    float*       oc = out + ((size_t)b * Cc + c) * plane;

    // Only the extreme tile columns can touch the zero-padded halo in w.
    const bool leftOK  = (w0 != 0);            // wave-uniform
    const bool rightOK = (w0 + TW != Wc);      // wave-uniform

    for (int rr = 0; rr < 2; ++rr) {
        const int r = 2 * q + rr;              // row within tile (wave-uniform)
        const int h = h0 + r;

        float acc[4][8];                        // 4 chunks x 8 outputs = one 32-wide row
#pragma unroll
        for (int m = 0; m < 4; ++m)
#pragma unroll
            for (int j = 0; j < 8; ++j) acc[m][j] = 0.f;

#pragma unroll
        for (int hh = 0; hh < 5; ++hh) {
            const int row = h + hh - 2;
            if (row < 0 || row >= Hc) continue;          // wave-uniform skip (zero pad)

            // 40-float row window covering cols [w0-4, w0+36): 10 aligned b128 loads,
            // only the first/last segment is (uniformly) predicated.
            const float* xrow = xc + (size_t)row * Wc;
            float xw[40];
#pragma unroll
            for (int s = 0; s < 10; ++s) {
                v4f v = {0.f, 0.f, 0.f, 0.f};
                const bool ok = (s == 0) ? leftOK : (s == 9) ? rightOK : true;
                if (ok)
                    v = *(const v4f*)(xrow + (w0 - 4 + 4 * s));   // global_load_b128
                xw[4*s+0] = v.x; xw[4*s+1] = v.y;
                xw[4*s+2] = v.z; xw[4*s+3] = v.w;
            }

#pragma unroll
            for (int ww = 0; ww < 5; ++ww) {
                const int k = hh * 5 + ww;               // matches reference tap order
#pragma unroll
                for (int m = 0; m < 4; ++m) {            // 8-wide output chunk
                    const float* wp = &wlds[(k * TH + r) * TW + 8 * m];
                    v4f wa = *(const v4f*)(wp);          // ds_load_b128
                    v4f wb = *(const v4f*)(wp + 4);      // ds_load_b128
                    const int base = 8 * m + ww + 2;     // window pos of output j=0
                    acc[m][0] = fmaf(xw[base + 0], wa.x, acc[m][0]);
                    acc[m][1] = fmaf(xw[base + 1], wa.y, acc[m][1]);
                    acc[m][2] = fmaf(xw[base + 2], wa.z, acc[m][2]);
                    acc[m][3] = fmaf(xw[base + 3], wa.w, acc[m][3]);
                    acc[m][4] = fmaf(xw[base + 4], wb.x, acc[m][4]);
                    acc[m][5] = fmaf(xw[base + 5], wb.y, acc[m][5]);
                    acc[m][6] = fmaf(xw[base + 6], wb.z, acc[m][6]);
                    acc[m][7] = fmaf(xw[base + 7], wb.w, acc[m][7]);
                }
            }
        }

        // Non-temporal stores: keep x/weights resident in the 192 MB L2.
        float* orow = oc + (size_t)h * Wc + w0;
#pragma unroll
        for (int m = 0; m < 4; ++m) {
            v4f o0 = {acc[m][0], acc[m][1], acc[m][2], acc[m][3]};
            v4f o1 = {acc[m][4], acc[m][5], acc[m][6], acc[m][7]};
            __builtin_nontemporal_store(o0, (v4f*)(orow + 8 * m));
            __builtin_nontemporal_store(o1, (v4f*)(orow + 8 * m + 4));
        }
    }
}

extern "C" void kernel_launch(void* const* d_in, const int* in_sizes, int n_in,
                              void* d_out, int out_size, void* d_ws, size_t ws_size,
                              hipStream_t stream) {
    const float* x   = (const float*)d_in[0];  // [8][64][256][256]
    const float* wts = (const float*)d_in[1];  // [25][8][1][256][256]
    float* out = (float*)d_out;                // [8][64][256][256]

    dim3 grid(Wc / TW, Hc / TH, Bc);           // (8, 32, 8) = 2048 blocks
    dim3 block(256);                           // 8 waves (wave32)
    dynconv5x5_kernel<<<grid, block, 0, stream>>>(x, wts, out);
}